// AugmentPipe_64862596104799
// MI455X (gfx1250) — compile-verified
//
#include <hip/hip_runtime.h>
#include <hip/hip_bf16.h>

// ---------------------------------------------------------------------------
// AugmentPipe (3D): fused warp + color(WMMA) + noise + cutout for gfx1250.
// images: [16,3,64,64,64] f32.  One fused pass => ~400MB HBM traffic (~20us
// floor at 23.3 TB/s).  Color matrix applied via V_WMMA_F32_16X16X4_F32.
// Per-voxel noise uses a cheap pcg4d counter hash (RNG must not make this
// bandwidth-bound kernel VALU-bound); one-shot setup kernel uses threefry.
// ---------------------------------------------------------------------------

typedef __attribute__((ext_vector_type(2))) float v2f;
typedef __attribute__((ext_vector_type(8))) float v8f;

#if defined(__AMDGCN__) && __has_builtin(__builtin_amdgcn_wmma_f32_16x16x4_f32)
#define USE_WMMA_F32X4 1
#else
#define USE_WMMA_F32X4 0
#endif

#define NB   16
#define NC   3
#define DV   64
#define VOX  (DV * DV * DV)            // 262144
#define WS_PER_B 36                    // 16 Ginv + 16 Cm + sigma + center xyz

// ------------------------------ RNG ---------------------------------------
__device__ __forceinline__ unsigned rotl32(unsigned x, int n) {
    return (x << n) | (x >> (32 - n));
}

// Threefry2x32 (20 rounds) — used only in the one-shot parameter setup.
__device__ __forceinline__ void threefry2x32(unsigned k0, unsigned k1,
                                             unsigned c0, unsigned c1,
                                             unsigned& o0, unsigned& o1) {
    const int R[8] = {13, 15, 26, 6, 17, 29, 16, 24};
    unsigned ks[3] = {k0, k1, 0x1BD11BDAu ^ k0 ^ k1};
    unsigned x0 = c0 + ks[0];
    unsigned x1 = c1 + ks[1];
#pragma unroll
    for (int j = 0; j < 5; ++j) {
#pragma unroll
        for (int r = 0; r < 4; ++r) {
            x0 += x1;
            x1 = rotl32(x1, R[(j & 1) * 4 + r]);
            x1 ^= x0;
        }
        x0 += ks[(j + 1) % 3];
        x1 += ks[(j + 2) % 3] + (unsigned)(j + 1);
    }
    o0 = x0; o1 = x1;
}

// pcg4d counter hash: 128 well-mixed bits in ~24 VALU ops (per-voxel path).
__device__ __forceinline__ void pcg4d(unsigned& a, unsigned& b,
                                      unsigned& c, unsigned& d) {
    a = a * 1664525u + 1013904223u;
    b = b * 1664525u + 1013904223u;
    c = c * 1664525u + 1013904223u;
    d = d * 1664525u + 1013904223u;
    a += b * d; b += c * a; c += a * b; d += b * c;
    a ^= a >> 16; b ^= b >> 16; c ^= c >> 16; d ^= d >> 16;
    a += b * d; b += c * a; c += a * b; d += b * c;
}

__device__ __forceinline__ float bits_to_unif(unsigned u) {
    return (float)(u >> 8) * (1.0f / 16777216.0f);   // [0,1)
}

__device__ __forceinline__ void boxmuller(unsigned a, unsigned b,
                                          float& n0, float& n1) {
    float u1 = fmaxf(bits_to_unif(a), 5.96e-8f);
    float u2 = bits_to_unif(b);
    float rad = sqrtf(-2.0f * __logf(u1));
    float ang = 6.28318530718f * u2;
    n0 = rad * __cosf(ang);
    n1 = rad * __sinf(ang);
}

// Single-output Box-Muller (skips the v_sin of the unused pair member).
__device__ __forceinline__ float boxmuller1(unsigned a, unsigned b) {
    float u1 = fmaxf(bits_to_unif(a), 5.96e-8f);
    float u2 = bits_to_unif(b);
    return sqrtf(-2.0f * __logf(u1)) * __cosf(6.28318530718f * u2);
}

__device__ __forceinline__ float tf_unif(unsigned stream, unsigned b, unsigned slot) {
    unsigned o0, o1;
    threefry2x32(42u, stream, b, slot, o0, o1);
    return bits_to_unif(o0);
}

__device__ __forceinline__ void tf_norm2(unsigned stream, unsigned b, unsigned slot,
                                         float& n0, float& n1) {
    unsigned o0, o1;
    threefry2x32(42u, stream, b, slot, o0, o1);
    boxmuller(o0, o1, n0, n1);
}

// --------------------------- 4x4 matrix helpers ----------------------------
struct M4 { float a[16]; };

__device__ __forceinline__ M4 m4_ident() {
    M4 M;
#pragma unroll
    for (int i = 0; i < 16; ++i) M.a[i] = ((i % 5) == 0) ? 1.f : 0.f;
    return M;
}
__device__ __forceinline__ M4 m4_mul(const M4& A, const M4& B) {
    M4 C;
#pragma unroll
    for (int i = 0; i < 4; ++i)
#pragma unroll
        for (int j = 0; j < 4; ++j) {
            float s = 0.f;
#pragma unroll
            for (int k = 0; k < 4; ++k) s += A.a[i * 4 + k] * B.a[k * 4 + j];
            C.a[i * 4 + j] = s;
        }
    return C;
}
__device__ __forceinline__ M4 m4_scale(float sx, float sy, float sz) {
    M4 M = m4_ident(); M.a[0] = sx; M.a[5] = sy; M.a[10] = sz; return M;
}
__device__ __forceinline__ M4 m4_translate(float tx, float ty, float tz) {
    M4 M = m4_ident(); M.a[3] = tx; M.a[7] = ty; M.a[11] = tz; return M;
}
__device__ __forceinline__ M4 m4_rotz(float t) {
    M4 M = m4_ident();
    float c = __cosf(t), s = __sinf(t);
    M.a[0] = c; M.a[1] = -s; M.a[4] = s; M.a[5] = c;
    return M;
}
__device__ __forceinline__ M4 m4_rotluma(float t) {  // Rodrigues about (1,1,1)/sqrt3
    const float aa = 0.5773502692f, a2 = 1.f / 3.f;
    float c = __cosf(t), s = __sinf(t), cc = 1.f - c;
    float d = a2 * cc + c, m = a2 * cc - aa * s, p = a2 * cc + aa * s;
    M4 M = m4_ident();
    M.a[0] = d; M.a[1] = m; M.a[2]  = p;
    M.a[4] = p; M.a[5] = d; M.a[6]  = m;
    M.a[8] = m; M.a[9] = p; M.a[10] = d;
    return M;
}
__device__ __forceinline__ M4 m4_lumaflip(float i) {   // I - 2*vv*i
    M4 M = m4_ident();
#pragma unroll
    for (int r = 0; r < 3; ++r)
#pragma unroll
        for (int c = 0; c < 3; ++c) M.a[r * 4 + c] -= (2.f / 3.f) * i;
    return M;
}
__device__ __forceinline__ M4 m4_sat(float s) {        // vv + (I - vv)*s
    M4 M;
#pragma unroll
    for (int i = 0; i < 16; ++i) M.a[i] = 0.f;
#pragma unroll
    for (int r = 0; r < 3; ++r)
#pragma unroll
        for (int c = 0; c < 3; ++c) {
            float vv = 1.f / 3.f;
            float id = (r == c) ? 1.f : 0.f;
            M.a[r * 4 + c] = vv + (id - vv) * s;
        }
    M.a[15] = s;
    return M;
}

// ------------------------ per-batch parameter setup ------------------------
__global__ void aug_setup_kernel(float* __restrict__ ws) {
    int b = threadIdx.x;
    if (b >= NB) return;
    float n0, n1, nd;

    // ---- geometry: G composed left-to-right (already the inverse map) ----
    M4 G = m4_ident();
    float u = tf_unif(0, b, 0);                          // xflip
    G = m4_mul(G, m4_scale(1.f - 2.f * floorf(u * 2.f), 1.f, 1.f));
    u = tf_unif(1, b, 0);                                // rotate90 about z
    G = m4_mul(G, m4_rotz(-1.5707963268f * floorf(u * 4.f)));
    {                                                    // integer translation
        float tx = (tf_unif(2, b, 0) * 2.f - 1.f) * 0.125f;
        float ty = (tf_unif(2, b, 1) * 2.f - 1.f) * 0.125f;
        float tz = (tf_unif(2, b, 2) * 2.f - 1.f) * 0.125f;
        G = m4_mul(G, m4_translate(-roundf(tx * DV), -roundf(ty * DV), -roundf(tz * DV)));
    }
    tf_norm2(3, b, 0, n0, n1);                           // isotropic scale (inv)
    { float s = exp2f(n0 * 0.2f); G = m4_mul(G, m4_scale(1.f / s, 1.f / s, 1.f / s)); }
    u = tf_unif(4, b, 0);                                // pre-rotation
    G = m4_mul(G, m4_rotz(-((u * 2.f - 1.f) * 3.14159265f)));
    tf_norm2(5, b, 0, n0, n1);                           // anisotropic scale (inv)
    { float s = exp2f(n0 * 0.2f); G = m4_mul(G, m4_scale(1.f / s, s, 1.f)); }
    u = tf_unif(6, b, 0);                                // post-rotation
    G = m4_mul(G, m4_rotz(-((u * 2.f - 1.f) * 3.14159265f)));
    {                                                    // fractional translation
        float t0, t1, t2;
        tf_norm2(7, b, 0, t0, t1);
        tf_norm2(7, b, 1, t2, nd);
        t0 *= 0.125f; t1 *= 0.125f; t2 *= 0.125f;
        G = m4_mul(G, m4_translate(-t0 * DV, -t1 * DV, -t2 * DV));
    }
    float* P = ws + b * WS_PER_B;
#pragma unroll
    for (int i = 0; i < 16; ++i) P[i] = G.a[i];

    // ---- color: Cm composed by left-multiplication ----
    M4 Cm = m4_ident();
    tf_norm2(8, b, 0, n0, n1);                           // brightness
    Cm = m4_mul(m4_translate(n0 * 0.2f, n0 * 0.2f, n0 * 0.2f), Cm);
    tf_norm2(9, b, 0, n0, n1);                           // contrast
    { float c = exp2f(n0 * 0.5f); Cm = m4_mul(m4_scale(c, c, c), Cm); }
    u = tf_unif(10, b, 0);                               // lumaflip
    Cm = m4_mul(m4_lumaflip(floorf(u * 2.f)), Cm);
    u = tf_unif(11, b, 0);                               // hue rotation
    Cm = m4_mul(m4_rotluma((u * 2.f - 1.f) * 3.14159265f), Cm);
    tf_norm2(12, b, 0, n0, n1);                          // saturation
    Cm = m4_mul(m4_sat(exp2f(n0 * 1.0f)), Cm);
#pragma unroll
    for (int i = 0; i < 16; ++i) P[16 + i] = Cm.a[i];

    tf_norm2(13, b, 0, n0, n1);                          // noise sigma
    P[32] = fabsf(n0) * 0.1f;
    P[33] = tf_unif(15, b, 0);                           // cutout center
    P[34] = tf_unif(15, b, 1);
    P[35] = tf_unif(15, b, 2);
}

// ------------------------------ fused main ---------------------------------
__global__ void __launch_bounds__(256)
aug_main_kernel(const float* __restrict__ img, float* __restrict__ out,
                const float* __restrict__ ws) {
    const int b   = blockIdx.y;
    const int vid = blockIdx.x * 256 + (int)threadIdx.x;    // 0..VOX-1
    const int x = vid & 63, y = (vid >> 6) & 63, z = vid >> 12;

    const float* P = ws + b * WS_PER_B;
    float g[12];
#pragma unroll
    for (int i = 0; i < 12; ++i) g[i] = P[i];

    // ---- inverse-map source coordinates (image-centered pixels) ----
    const float cx = (float)x - 31.5f, cy = (float)y - 31.5f, cz = (float)z - 31.5f;
    float sxf = g[0] * cx + g[1] * cy + g[2]  * cz + g[3]  + 31.5f;
    float syf = g[4] * cx + g[5] * cy + g[6]  * cz + g[7]  + 31.5f;
    float szf = g[8] * cx + g[9] * cy + g[10] * cz + g[11] + 31.5f;

    float xf = floorf(sxf), yf = floorf(syf), zf = floorf(szf);
    float wx = sxf - xf,    wy = syf - yf,    wz = szf - zf;
    int ix0 = (int)xf, iy0 = (int)yf, iz0 = (int)zf;
    int ix1 = ix0 + 1, iy1 = iy0 + 1, iz1 = iz0 + 1;

    // validity folded into weights (zero padding), addresses clamped (safe)
    float ax0 = (1.f - wx) * (((unsigned)ix0 < 64u) ? 1.f : 0.f);
    float ax1 = wx         * (((unsigned)ix1 < 64u) ? 1.f : 0.f);
    float ay0 = (1.f - wy) * (((unsigned)iy0 < 64u) ? 1.f : 0.f);
    float ay1 = wy         * (((unsigned)iy1 < 64u) ? 1.f : 0.f);
    float az0 = (1.f - wz) * (((unsigned)iz0 < 64u) ? 1.f : 0.f);
    float az1 = wz         * (((unsigned)iz1 < 64u) ? 1.f : 0.f);

    int jx0 = min(max(ix0, 0), 63), jx1 = min(max(ix1, 0), 63);
    int jy0 = min(max(iy0, 0), 63), jy1 = min(max(iy1, 0), 63);
    int jz0 = min(max(iz0, 0), 63), jz1 = min(max(iz1, 0), 63);
    int oz0 = jz0 << 12, oz1 = jz1 << 12, oy0 = jy0 << 6, oy1 = jy1 << 6;
    int o000 = oz0 + oy0 + jx0, o001 = oz0 + oy0 + jx1;
    int o010 = oz0 + oy1 + jx0, o011 = oz0 + oy1 + jx1;
    int o100 = oz1 + oy0 + jx0, o101 = oz1 + oy0 + jx1;
    int o110 = oz1 + oy1 + jx0, o111 = oz1 + oy1 + jx1;
    float w000 = az0 * ay0 * ax0, w001 = az0 * ay0 * ax1;
    float w010 = az0 * ay1 * ax0, w011 = az0 * ay1 * ax1;
    float w100 = az1 * ay0 * ax0, w101 = az1 * ay0 * ax1;
    float w110 = az1 * ay1 * ax0, w111 = az1 * ay1 * ax1;

    const float* base = img + (size_t)b * NC * VOX;
    float ch[3];
#pragma unroll
    for (int c = 0; c < 3; ++c) {
        const float* p = base + (size_t)c * VOX;
        ch[c] = w000 * p[o000] + w001 * p[o001] + w010 * p[o010] + w011 * p[o011]
              + w100 * p[o100] + w101 * p[o101] + w110 * p[o110] + w111 * p[o111];
    }
    float r = ch[0], gg = ch[1], bb = ch[2];

    // ---- color transform: D(3xN) = Cm(3x4) x [r,g,b,1](4xN) ----
    const float* Cmp = P + 16;
    float R, Gc, Bc;
#if USE_WMMA_F32X4
    {
        const int lane = (int)(threadIdx.x & 31u);
        const int mrow = lane & 15;
        // A operand (16x4 f32): lanes 0-15 carry K=0,1 of row M=lane;
        // lanes 16-31 carry K=2,3.  Rows >= 3 are zero.
        const int mr = (mrow < 3) ? mrow : 0;
        const float rowmask = (mrow < 3) ? 1.f : 0.f;
        const int col0 = (lane < 16) ? 0 : 2;
        v2f A;
        A.x = Cmp[mr * 4 + col0]     * rowmask;
        A.y = Cmp[mr * 4 + col0 + 1] * rowmask;

        // B operand (4x16 f32): column n = voxel of one lane; K=0,1 in lanes
        // 0-15, K=2,3 in lanes 16-31.  Two WMMAs cover the 32 lanes' voxels.
        const float bbx = __shfl_xor(bb, 16);
        const float rx  = __shfl_xor(r, 16);
        const float gx  = __shfl_xor(gg, 16);
        v2f B1, B2;
        B1.x = (lane < 16) ? r  : bbx;  B1.y = (lane < 16) ? gg : 1.f;
        B2.x = (lane < 16) ? rx : bb;   B2.y = (lane < 16) ? gx : 1.f;

        v8f Cz = {0.f, 0.f, 0.f, 0.f, 0.f, 0.f, 0.f, 0.f};
        v8f D1 = __builtin_amdgcn_wmma_f32_16x16x4_f32(
            false, A, false, B1, (short)0, Cz, false, false);
        v8f D2 = __builtin_amdgcn_wmma_f32_16x16x4_f32(
            false, A, false, B2, (short)0, Cz, false, false);

        // D layout: channel m of column n lives in VGPR m of lane n (n<16).
        const float t0 = __shfl_xor(D2[0], 16);
        const float t1 = __shfl_xor(D2[1], 16);
        const float t2 = __shfl_xor(D2[2], 16);
        R  = (lane < 16) ? D1[0] : t0;
        Gc = (lane < 16) ? D1[1] : t1;
        Bc = (lane < 16) ? D1[2] : t2;
    }
#else
    R  = Cmp[0] * r + Cmp[1] * gg + Cmp[2]  * bb + Cmp[3];
    Gc = Cmp[4] * r + Cmp[5] * gg + Cmp[6]  * bb + Cmp[7];
    Bc = Cmp[8] * r + Cmp[9] * gg + Cmp[10] * bb + Cmp[11];
#endif

    // ---- per-element Gaussian noise (cheap pcg4d counter hash) ----
    unsigned qa = (unsigned)vid;
    unsigned qb = (unsigned)b ^ 0xCAFEF00Du;
    unsigned qc = 0x9E3779B9u;
    unsigned qd = 42u;
    pcg4d(qa, qb, qc, qd);
    float n0, n1;
    boxmuller(qa, qb, n0, n1);
    const float n2 = boxmuller1(qc, qd);
    const float sig = P[32];
    R  += n0 * sig;
    Gc += n1 * sig;
    Bc += n2 * sig;

    // ---- cutout box mask ----
    const float fx = ((float)x + 0.5f) * (1.f / 64.f);
    const float fy = ((float)y + 0.5f) * (1.f / 64.f);
    const float fz = ((float)z + 0.5f) * (1.f / 64.f);
    const bool keep = (fabsf(fx - P[33]) >= 0.25f) |
                      (fabsf(fy - P[34]) >= 0.25f) |
                      (fabsf(fz - P[35]) >= 0.25f);
    const float msk = keep ? 1.f : 0.f;

    float* ob = out + (size_t)b * NC * VOX;
    ob[vid]            = R  * msk;
    ob[VOX + vid]      = Gc * msk;
    ob[2 * VOX + vid]  = Bc * msk;
}

// ------------------------------- launcher ----------------------------------
extern "C" void kernel_launch(void* const* d_in, const int* in_sizes, int n_in,
                              void* d_out, int out_size, void* d_ws, size_t ws_size,
                              hipStream_t stream) {
    (void)in_sizes; (void)n_in; (void)out_size; (void)ws_size;
    const float* img = (const float*)d_in[0];
    float*       out = (float*)d_out;
    float*       ws  = (float*)d_ws;      // needs NB*WS_PER_B*4 = 2304 bytes

    aug_setup_kernel<<<1, 32, 0, stream>>>(ws);
    dim3 grid(VOX / 256, NB);
    aug_main_kernel<<<grid, 256, 0, stream>>>(img, out, ws);
}